// PairwiseSoftDTW_40003325395195
// MI455X (gfx1250) — compile-verified
//
#include <hip/hip_runtime.h>

typedef __attribute__((ext_vector_type(2))) float v2f;
typedef __attribute__((ext_vector_type(8))) float v8f;

#define TT   128       // sequence length
#define DIMS 8
#define NA   128
#define NB   16
#define BIGF 1e10f

// One workgroup per (a,b) pair. 128 threads = 4 wave32.
// Phase 1: D[i][j] = |X[a][i]|^2 + |Y[b][j]|^2 - 2 * (X[a] · Y[b]^T)[i][j]
//          cross term via V_WMMA_F32_16X16X4_F32, D kept in LDS (64 KB).
// Phase 2: soft-DTW anti-diagonal scan fully in LDS, one barrier per step.
__global__ __launch_bounds__(128)
void softdtw_fused_kernel(const float* __restrict__ X,
                          const float* __restrict__ Y,
                          float* __restrict__ out)
{
    __shared__ float Dlds[TT * TT];   // 64 KB distance tile
    __shared__ float sq[2 * TT];      // [0..127]=|X row|^2, [128..255]=|Y row|^2
    __shared__ float rows[3 * TT];    // rotating DP anti-diagonals

    const int a    = blockIdx.y;
    const int b    = blockIdx.x;
    const int tid  = threadIdx.x;
    const int lane = tid & 31;
    const int wave = tid >> 5;
    const int half = lane >> 4;   // 0: lanes 0-15, 1: lanes 16-31
    const int lr   = lane & 15;

    const float* __restrict__ Xa = X + (size_t)a * TT * DIMS;
    const float* __restrict__ Yb = Y + (size_t)b * TT * DIMS;

    // ---- squared norms (one row per thread) ----
    {
        float sx = 0.f, sy = 0.f;
#pragma unroll
        for (int d = 0; d < DIMS; ++d) {
            float xv = Xa[tid * DIMS + d];
            float yv = Yb[tid * DIMS + d];
            sx = fmaf(xv, xv, sx);
            sy = fmaf(yv, yv, sy);
        }
        sq[tid]      = sx;
        sq[TT + tid] = sy;
    }
    __syncthreads();

    // ---- cross term via WMMA: wave w owns tile-rows mt = 2w, 2w+1; nt = 0..7 ----
    // A-frag (16x4 f32): lane half h -> VGPR0 = A[M][2h], VGPR1 = A[M][2h+1], M = lr
    // B-frag (4x16)    : lane half h -> VGPR0 = B[2h][n] = Y[n][2h], n = lr   (B = Y-tile^T)
    for (int mti = 0; mti < 2; ++mti) {
        const int mt = 2 * wave + mti;
        const float* xr = Xa + (mt * 16 + lr) * DIMS + 2 * half;
        v2f a_lo, a_hi;
        a_lo.x = xr[0]; a_lo.y = xr[1];   // K = 0..3 across halves
        a_hi.x = xr[4]; a_hi.y = xr[5];   // K = 4..7 across halves
        for (int nt = 0; nt < 8; ++nt) {
            const float* yr = Yb + (nt * 16 + lr) * DIMS + 2 * half;
            v2f b_lo, b_hi;
            b_lo.x = yr[0]; b_lo.y = yr[1];
            b_hi.x = yr[4]; b_hi.y = yr[5];
            v8f c = {};
            c = __builtin_amdgcn_wmma_f32_16x16x4_f32(false, a_lo, false, b_lo,
                                                      (short)0, c, false, false);
            c = __builtin_amdgcn_wmma_f32_16x16x4_f32(false, a_hi, false, b_hi,
                                                      (short)0, c, false, false);
            // C/D layout: VGPR r -> M = r + 8*half, N = lr
#pragma unroll
            for (int r = 0; r < 8; ++r) {
                const int i = mt * 16 + half * 8 + r;
                const int j = nt * 16 + lr;
                Dlds[i * TT + j] = sq[i] + sq[TT + j] - 2.0f * c[r];
            }
        }
    }

    // ---- soft-DTW anti-diagonal DP (gamma = 1) ----
    rows[tid]      = BIGF;   // prev2
    rows[TT + tid] = BIGF;   // prev1
    __syncthreads();

    int ip2 = 0, ip1 = 1, icur = 2;
    const int i = tid;
#pragma unroll 1
    for (int p = 0; p < 2 * TT - 1; ++p) {
        const int  j     = p - i;
        const bool valid = (j >= 0) && (j < TT);
        const float d    = valid ? Dlds[i * TT + j] : BIGF;
        const float diag = (i == 0) ? ((p == 0) ? 0.0f : BIGF) : rows[ip2 * TT + i - 1];
        const float up   = (i == 0) ? BIGF : rows[ip1 * TT + i - 1];
        const float left = rows[ip1 * TT + i];
        const float r0 = -diag, r1 = -up, r2 = -left;
        const float rmax = fmaxf(fmaxf(r0, r1), r2);
        const float rsum = __expf(r0 - rmax) + __expf(r1 - rmax) + __expf(r2 - rmax);
        const float softmin = -(__logf(rsum) + rmax);
        rows[icur * TT + i] = valid ? (d + softmin) : BIGF;
        __syncthreads();
        const int t = ip2; ip2 = ip1; ip1 = icur; icur = t;
    }

    if (tid == TT - 1)
        out[a * NB + b] = rows[ip1 * TT + (TT - 1)];
}

extern "C" void kernel_launch(void* const* d_in, const int* in_sizes, int n_in,
                              void* d_out, int out_size, void* d_ws, size_t ws_size,
                              hipStream_t stream) {
    (void)in_sizes; (void)n_in; (void)out_size; (void)d_ws; (void)ws_size;
    const float* X = (const float*)d_in[0];   // (128, 128, 8) f32
    const float* Y = (const float*)d_in[1];   // (16, 128, 8)  f32
    float* out = (float*)d_out;               // (128, 16)     f32
    dim3 grid(NB, NA);
    softdtw_fused_kernel<<<grid, TT, 0, stream>>>(X, Y, out);
}